// GAT_17824114278677
// MI455X (gfx1250) — compile-verified
//
#include <hip/hip_runtime.h>
#include <cstddef>

#define NEGV  (-9e15f)
#define LALPHA 0.2f

typedef float v2f __attribute__((ext_vector_type(2)));
typedef float v8f __attribute__((ext_vector_type(8)));

__device__ __forceinline__ float lrelu(float x) { return x > 0.f ? x : LALPHA * x; }
__device__ __forceinline__ float elu1(float x)  { return x > 0.f ? x : (__expf(x) - 1.f); }

// ---------------------------------------------------------------------------
// Projection GEMM: out[h] = X @ W[h]   (X: [N,ldx] row-major, W[h]: [K, FT*16])
// One wave computes 16 rows x FT*16 cols with FT accumulators; the A fragment
// (16x4 of X) is loaded once per K-step and reused across the FT column WMMAs.
// A frag: lane M = lane&15; half-wave picks K pair {0,1}/{2,3}.
// B frag: lane N = lane&15; half-wave picks K pair {0,1}/{2,3}.
// C     : vgpr r -> row r (lanes 0-15) / row 8+r (lanes 16-31).
// ---------------------------------------------------------------------------
template <int FT>
__global__ void gat_proj(const float* __restrict__ X, const float* __restrict__ Wt,
                         float* __restrict__ out, int N, int K, int ldx) {
  constexpr int F = FT * 16;
  const int tm = blockIdx.x, h = blockIdx.z;
  const float* Wm = Wt + (size_t)h * K * F;
  float* O = out + (size_t)h * N * F;
  const int lane = threadIdx.x & 31;
  const int rc = lane & 15;
  const int kh = (lane >> 4) << 1;
  const int i = tm * 16 + rc;
  v8f c[FT];
#pragma unroll
  for (int ft = 0; ft < FT; ++ft) c[ft] = {};
  for (int k0 = 0; k0 < K; k0 += 4) {
    const int ks = k0 + kh;
    v2f a;
    a.x = X[(size_t)i * ldx + ks];
    a.y = X[(size_t)i * ldx + ks + 1];
#pragma unroll
    for (int ft = 0; ft < FT; ++ft) {
      v2f b;
      const float* wp = Wm + (size_t)ks * F + ft * 16 + rc;
      b.x = wp[0];
      b.y = wp[F];
      c[ft] = __builtin_amdgcn_wmma_f32_16x16x4_f32(false, a, false, b, (short)0,
                                                    c[ft], false, false);
    }
  }
  const int half = lane >> 4;
#pragma unroll
  for (int ft = 0; ft < FT; ++ft)
#pragma unroll
    for (int r = 0; r < 8; ++r) {
      const int row = tm * 16 + r + 8 * half;
      O[(size_t)row * F + ft * 16 + rc] = c[ft][r];
    }
}

// ---------------------------------------------------------------------------
// Attention coefficients: fsrc[h,i] = Wh[h,i,:]·a[h,:F], fdst = Wh·a[h,F:2F]
// ---------------------------------------------------------------------------
__global__ void gat_coef(const float* __restrict__ Wh, const float* __restrict__ a,
                         float* __restrict__ fsrc, float* __restrict__ fdst,
                         int N, int F) {
  const int t = blockIdx.x * blockDim.x + threadIdx.x;
  const int h = t / N, i = t % N;
  const float* w  = Wh + ((size_t)h * N + i) * F;
  const float* av = a + (size_t)h * 2 * F;
  float s0 = 0.f, s1 = 0.f;
  for (int k = 0; k < F; ++k) {
    const float v = w[k];
    s0 += v * av[k];
    s1 += v * av[F + k];
  }
  fsrc[(size_t)h * N + i] = s0;
  fdst[(size_t)h * N + i] = s1;
}

// ---------------------------------------------------------------------------
// Row softmax stats: one block per adjacency row; all heads in one adj pass.
// m[h,i] = max_j e(h,i,j),  s[h,i] = sum_j exp(e - m). Masked entries e=NEG.
// ---------------------------------------------------------------------------
template <int HEADS>
__global__ void gat_stats(const float* __restrict__ fsrc, const float* __restrict__ fdst,
                          const int* __restrict__ adj, float* __restrict__ m,
                          float* __restrict__ s, int N) {
  __shared__ float red[HEADS][256];
  const int i = blockIdx.x, tid = threadIdx.x;
  const int* arow = adj + (size_t)i * N;
  float fs[HEADS], mx[HEADS];
#pragma unroll
  for (int h = 0; h < HEADS; ++h) { fs[h] = fsrc[(size_t)h * N + i]; mx[h] = NEGV; }
  for (int j = tid; j < N; j += 256) {
    const int av = arow[j];
#pragma unroll
    for (int h = 0; h < HEADS; ++h) {
      const float e = (av > 0) ? lrelu(fs[h] + fdst[(size_t)h * N + j]) : NEGV;
      mx[h] = fmaxf(mx[h], e);
    }
  }
#pragma unroll
  for (int h = 0; h < HEADS; ++h) red[h][tid] = mx[h];
  __syncthreads();
  for (int off = 128; off > 0; off >>= 1) {
    if (tid < off)
#pragma unroll
      for (int h = 0; h < HEADS; ++h)
        red[h][tid] = fmaxf(red[h][tid], red[h][tid + off]);
    __syncthreads();
  }
  float mf[HEADS];
#pragma unroll
  for (int h = 0; h < HEADS; ++h) mf[h] = red[h][0];
  __syncthreads();
  float sm[HEADS];
#pragma unroll
  for (int h = 0; h < HEADS; ++h) sm[h] = 0.f;
  for (int j = tid; j < N; j += 256) {
    const int av = arow[j];
#pragma unroll
    for (int h = 0; h < HEADS; ++h) {
      const float e = (av > 0) ? lrelu(fs[h] + fdst[(size_t)h * N + j]) : NEGV;
      sm[h] += __expf(e - mf[h]);
    }
  }
#pragma unroll
  for (int h = 0; h < HEADS; ++h) red[h][tid] = sm[h];
  __syncthreads();
  for (int off = 128; off > 0; off >>= 1) {
    if (tid < off)
#pragma unroll
      for (int h = 0; h < HEADS; ++h) red[h][tid] += red[h][tid + off];
    __syncthreads();
  }
  if (tid == 0)
#pragma unroll
    for (int h = 0; h < HEADS; ++h) {
      m[(size_t)h * N + i] = mf[h];
      s[(size_t)h * N + i] = red[h][0];
    }
}

// ---------------------------------------------------------------------------
// Attention * V, flash-style. Block = CT waves covering RT*16 rows x CT*16
// cols of out[h] = elu( softmax(e[h]) @ Wh[h] ). The RT*16 x 64 tile of
// exp(e - m) is staged once in LDS (stride 68 -> conflict-free A-frag reads)
// and shared by all CT column-waves; each B fragment (Wh) is loaded once and
// reused across the RT row-strip WMMAs. The next K-chunk of adj is prefetched
// (global_prefetch_b8) while the current chunk is consumed.
// ---------------------------------------------------------------------------
template <int CT, int RT>
__global__ void gat_av(const float* __restrict__ fsrc, const float* __restrict__ fdst,
                       const float* __restrict__ m, const float* __restrict__ s,
                       const int* __restrict__ adj, const float* __restrict__ Wh,
                       float* __restrict__ out, int ldout, int N) {
  constexpr int F = CT * 16;
  constexpr int ROWS = RT * 16;
  __shared__ float P[ROWS][68];
  __shared__ float rf[ROWS], rm[ROWS], rs[ROWS];
  const int h = blockIdx.y;
  const int i0 = blockIdx.x * ROWS;
  const float* fd = fdst + (size_t)h * N;
  const float* W = Wh + (size_t)h * N * F;
  const int tid = threadIdx.x;
  for (int r = tid; r < ROWS; r += CT * 32) {
    rf[r] = fsrc[(size_t)h * N + i0 + r];
    rm[r] = m[(size_t)h * N + i0 + r];
    rs[r] = 1.f / s[(size_t)h * N + i0 + r];
  }
  const int lane = tid & 31;
  const int wave = tid >> 5;
  const int n0 = wave * 16;
  const int rc = lane & 15;
  const int kh = (lane >> 4) << 1;
  v8f c[RT];
#pragma unroll
  for (int rt = 0; rt < RT; ++rt) c[rt] = {};
  for (int k0 = 0; k0 < N; k0 += 64) {
    __syncthreads();
    // stage exp(e - m): ROWS x 64 K chunk, computed once per block
    for (int idx = tid; idx < ROWS * 64; idx += CT * 32) {
      const int r = idx >> 6, cc = idx & 63;
      const int j = k0 + cc;
      const size_t aoff = (size_t)(i0 + r) * N + j;
      const int av = adj[aoff];
      if (k0 + 64 < N) __builtin_prefetch(&adj[aoff + 64], 0, 1);
      const float e = (av > 0) ? lrelu(rf[r] + fd[j]) : NEGV;
      P[r][cc] = __expf(e - rm[r]);
    }
    __syncthreads();
#pragma unroll
    for (int kk = 0; kk < 16; ++kk) {
      const int kb = 4 * kk + kh;
      v2f b;
      const float* wp = W + (size_t)(k0 + kb) * F + n0 + rc;
      b.x = wp[0];
      b.y = wp[F];
#pragma unroll
      for (int rt = 0; rt < RT; ++rt) {
        v2f a;
        a.x = P[rt * 16 + rc][kb];
        a.y = P[rt * 16 + rc][kb + 1];
        c[rt] = __builtin_amdgcn_wmma_f32_16x16x4_f32(false, a, false, b, (short)0,
                                                      c[rt], false, false);
      }
    }
  }
  const int half = lane >> 4;
#pragma unroll
  for (int rt = 0; rt < RT; ++rt)
#pragma unroll
    for (int r = 0; r < 8; ++r) {
      const int mr = rt * 16 + r + 8 * half;
      float v = c[rt][r] * rs[mr];
      v = elu1(v);
      out[(size_t)(i0 + mr) * ldout + h * F + n0 + rc] = v;
    }
}

// ---------------------------------------------------------------------------
extern "C" void kernel_launch(void* const* d_in, const int* in_sizes, int n_in,
                              void* d_out, int out_size, void* d_ws, size_t ws_size,
                              hipStream_t stream) {
  (void)in_sizes; (void)n_in; (void)out_size; (void)ws_size;
  const float* x   = (const float*)d_in[0];
  const int*   adj = (const int*)d_in[1];
  const float* W1  = (const float*)d_in[2];
  const float* a1  = (const float*)d_in[3];
  const float* W2  = (const float*)d_in[4];
  const float* a2  = (const float*)d_in[5];
  float* outp = (float*)d_out;

  const int N = 4096, FIN = 512, H = 8, HID = 64, OUT = 16;

  float* ws = (float*)d_ws;
  size_t off = 0;
  float* Wh1   = ws + off; off += (size_t)H * N * HID;   // 8 MB
  float* fsrc1 = ws + off; off += (size_t)H * N;
  float* fdst1 = ws + off; off += (size_t)H * N;
  float* m1    = ws + off; off += (size_t)H * N;
  float* s1    = ws + off; off += (size_t)H * N;
  float* hcat  = ws + off; off += (size_t)N * H * HID;   // 8 MB
  float* Wh2   = ws + off; off += (size_t)N * OUT;
  float* fsrc2 = ws + off; off += (size_t)N;
  float* fdst2 = ws + off; off += (size_t)N;
  float* m2    = ws + off; off += (size_t)N;
  float* s2    = ws + off; off += (size_t)N;

  // ---- Layer 1 (8 heads) ----
  gat_proj<4><<<dim3(N / 16, 1, H), 32, 0, stream>>>(x, W1, Wh1, N, FIN, FIN);
  gat_coef<<<(H * N) / 256, 256, 0, stream>>>(Wh1, a1, fsrc1, fdst1, N, HID);
  gat_stats<8><<<N, 256, 0, stream>>>(fsrc1, fdst1, adj, m1, s1, N);
  gat_av<4, 2><<<dim3(N / 32, H), 128, 0, stream>>>(fsrc1, fdst1, m1, s1, adj, Wh1,
                                                    hcat, H * HID, N);

  // ---- Layer 2 (single head) ----
  gat_proj<1><<<dim3(N / 16, 1, 1), 32, 0, stream>>>(hcat, W2, Wh2, N, H * HID,
                                                     H * HID);
  gat_coef<<<N / 256, 256, 0, stream>>>(Wh2, a2, fsrc2, fdst2, N, OUT);
  gat_stats<1><<<N, 256, 0, stream>>>(fsrc2, fdst2, adj, m2, s2, N);
  gat_av<1, 1><<<dim3(N / 16, 1), 32, 0, stream>>>(fsrc2, fdst2, m2, s2, adj, Wh2,
                                                   outp, OUT, N);
}